// OptimizedGraphSAGE_14121852469958
// MI455X (gfx1250) — compile-verified
//
#include <hip/hip_runtime.h>
#include <math.h>

#define N_NODES 100000
#define N_EDGES 1600000
#define D 128
#define C 40
#define BN_EPS 1e-5f

typedef __attribute__((ext_vector_type(2))) float v2f;
typedef __attribute__((ext_vector_type(8))) float v8f;

__device__ __forceinline__ void atomic_add_f32(float* p, float v) {
    __hip_atomic_fetch_add(p, v, __ATOMIC_RELAXED, __HIP_MEMORY_SCOPE_AGENT);
}

// ---------------------------------------------------------------- zero fill
__global__ void zero_kernel(float* __restrict__ p, size_t n) {
    size_t i = (size_t)blockIdx.x * blockDim.x + threadIdx.x;
    size_t stride = (size_t)gridDim.x * blockDim.x;
    for (; i < n; i += stride) p[i] = 0.0f;
}

// ---------------------------------------------------------------- weight transpose: outT[c*K + k] = in[k*Cd + c]
__global__ void transpose_kernel(const float* __restrict__ in, float* __restrict__ outT,
                                 int K, int Cd) {
    int i = blockIdx.x * blockDim.x + threadIdx.x;
    if (i < K * Cd) {
        int k = i / Cd;
        int c = i - k * Cd;
        outT[(size_t)c * K + k] = in[i];
    }
}

// ---------------------------------------------------------------- degrees
__global__ void degree_kernel(const long long* __restrict__ ei, float* __restrict__ deg) {
    int e = blockIdx.x * blockDim.x + threadIdx.x;
    if (e < N_EDGES) {
        int dst = (int)ei[(size_t)N_EDGES + e];
        atomic_add_f32(&deg[dst], 1.0f);
    }
}

__global__ void invdeg_kernel(float* __restrict__ deg) {
    int i = blockIdx.x * blockDim.x + threadIdx.x;
    if (i < N_NODES) deg[i] = 1.0f / fmaxf(deg[i], 1.0f);
}

// ---------------------------------------------------------------- scatter-add of features over edges
// one wave per edge, 4 features per lane (float4 gather, 4 f32 atomics)
__global__ __launch_bounds__(256) void scatter_kernel(const long long* __restrict__ ei,
                                                      const float* __restrict__ feat,
                                                      float* __restrict__ msg) {
    int t = blockIdx.x * 256 + threadIdx.x;
    int e = t >> 5;              // 32 lanes per edge
    int f = (t & 31) * 4;        // feature quad
    if (e < N_EDGES) {
        long long s = ei[e];
        long long d = ei[(size_t)N_EDGES + e];
        const float4 v = *(const float4*)(feat + (size_t)s * D + f);
        float* dp = msg + (size_t)d * D + f;
        atomic_add_f32(dp + 0, v.x);
        atomic_add_f32(dp + 1, v.y);
        atomic_add_f32(dp + 2, v.z);
        atomic_add_f32(dp + 3, v.w);
    }
}

// ---------------------------------------------------------------- fused dual-GEMM, H=128 outputs
// out = epilogue( (msg*invdeg) @ wl + bias + xin @ wr ), weights pre-transposed: wT[col][k]
// mode 0: relu                         (layer 0)
// mode 1: batchnorm, relu, + residual  (layer 1)
// Block: 256 threads = 8 waves; 16 output rows; wave w owns cols [16w, 16w+16).
// A/X tiles staged in LDS once per block (invdeg folded in at stage time).
__global__ __launch_bounds__(256) void gemm128_kernel(
    const float* __restrict__ msg, const float* __restrict__ invdeg,
    const float* __restrict__ xin,
    const float* __restrict__ wlT, const float* __restrict__ wrT,
    const float* __restrict__ bias,
    const float* __restrict__ gamma, const float* __restrict__ beta,
    const float* __restrict__ rmean, const float* __restrict__ rvar,
    const float* __restrict__ resid,
    float* __restrict__ out, int mode) {
    __shared__ float sA[16][D + 4];
    __shared__ float sX[16][D + 4];
    const int tid = threadIdx.x;
    const int base_row = blockIdx.x * 16;

    // stage 16x128 A (scaled) and X tiles: 512 float4 each, 2 per thread
    #pragma unroll
    for (int i = tid; i < 16 * (D / 4); i += 256) {
        const int r  = i >> 5;
        const int c4 = (i & 31) * 4;
        const int grow = base_row + r;
        const float inv = invdeg[grow];
        const float4 a  = *(const float4*)(msg + (size_t)grow * D + c4);
        const float4 xv = *(const float4*)(xin + (size_t)grow * D + c4);
        sA[r][c4 + 0] = a.x * inv;  sA[r][c4 + 1] = a.y * inv;
        sA[r][c4 + 2] = a.z * inv;  sA[r][c4 + 3] = a.w * inv;
        sX[r][c4 + 0] = xv.x;  sX[r][c4 + 1] = xv.y;
        sX[r][c4 + 2] = xv.z;  sX[r][c4 + 3] = xv.w;
    }
    __syncthreads();

    const int wave = tid >> 5;
    const int lane = tid & 31;
    const int half = lane >> 4;      // K half for A/B frags
    const int lm   = lane & 15;
    const int col  = wave * 16 + lm; // B/C column held by this lane
    const float* __restrict__ blrow = wlT + (size_t)col * D;
    const float* __restrict__ brrow = wrT + (size_t)col * D;

    v8f acc = {};
    #pragma unroll 4
    for (int k = 0; k < D; k += 4) {
        const int ka = k + 2 * half;
        v2f a0 = *(const v2f*)&sA[lm][ka];
        v2f b0 = *(const v2f*)(blrow + ka);
        acc = __builtin_amdgcn_wmma_f32_16x16x4_f32(false, a0, false, b0, (short)0, acc, false, false);
        v2f a1 = *(const v2f*)&sX[lm][ka];
        v2f b1 = *(const v2f*)(brrow + ka);
        acc = __builtin_amdgcn_wmma_f32_16x16x4_f32(false, a1, false, b1, (short)0, acc, false, false);
    }

    const float bc = bias[col];
    float g = 0.f, bt = 0.f, mu = 0.f, isd = 1.f;
    if (mode == 1) {
        g = gamma[col]; bt = beta[col]; mu = rmean[col];
        isd = rsqrtf(rvar[col] + BN_EPS);
    }
    #pragma unroll
    for (int i = 0; i < 8; ++i) {
        const int orow = base_row + i + 8 * half;  // C/D layout: M = i + 8*half
        float v = acc[i] + bc;
        if (mode == 1) {
            v = (v - mu) * isd * g + bt;
            v = fmaxf(v, 0.0f) + resid[(size_t)orow * D + col];
        } else {
            v = fmaxf(v, 0.0f);
        }
        out[(size_t)orow * D + col] = v;
    }
}

// ---------------------------------------------------------------- fused dual-GEMM, C=40 outputs (padded to 48)
// Block: 96 threads = 3 waves (col tiles 0..2); cols >= 40 masked. Weights pre-transposed [C][128].
__global__ __launch_bounds__(96) void gemm_out_kernel(
    const float* __restrict__ msg, const float* __restrict__ invdeg,
    const float* __restrict__ xin,
    const float* __restrict__ wlT, const float* __restrict__ wrT,
    const float* __restrict__ bias, float* __restrict__ out) {
    __shared__ float sA[16][D + 4];
    __shared__ float sX[16][D + 4];
    const int tid = threadIdx.x;
    const int base_row = blockIdx.x * 16;

    for (int i = tid; i < 16 * (D / 4); i += 96) {
        const int r  = i >> 5;
        const int c4 = (i & 31) * 4;
        const int grow = base_row + r;
        const float inv = invdeg[grow];
        const float4 a  = *(const float4*)(msg + (size_t)grow * D + c4);
        const float4 xv = *(const float4*)(xin + (size_t)grow * D + c4);
        sA[r][c4 + 0] = a.x * inv;  sA[r][c4 + 1] = a.y * inv;
        sA[r][c4 + 2] = a.z * inv;  sA[r][c4 + 3] = a.w * inv;
        sX[r][c4 + 0] = xv.x;  sX[r][c4 + 1] = xv.y;
        sX[r][c4 + 2] = xv.z;  sX[r][c4 + 3] = xv.w;
    }
    __syncthreads();

    const int wave = tid >> 5;
    const int lane = tid & 31;
    const int half = lane >> 4;
    const int lm   = lane & 15;
    const int col  = wave * 16 + lm;        // 0..47
    const bool cok = (col < C);
    const int colc = cok ? col : 0;         // clamp for safe (discarded) loads
    const float* __restrict__ blrow = wlT + (size_t)colc * D;
    const float* __restrict__ brrow = wrT + (size_t)colc * D;
    const float zmask = cok ? 1.0f : 0.0f;

    v8f acc = {};
    #pragma unroll 4
    for (int k = 0; k < D; k += 4) {
        const int ka = k + 2 * half;
        v2f a0 = *(const v2f*)&sA[lm][ka];
        v2f b0 = *(const v2f*)(blrow + ka);
        b0.x *= zmask; b0.y *= zmask;
        acc = __builtin_amdgcn_wmma_f32_16x16x4_f32(false, a0, false, b0, (short)0, acc, false, false);
        v2f a1 = *(const v2f*)&sX[lm][ka];
        v2f b1 = *(const v2f*)(brrow + ka);
        b1.x *= zmask; b1.y *= zmask;
        acc = __builtin_amdgcn_wmma_f32_16x16x4_f32(false, a1, false, b1, (short)0, acc, false, false);
    }
    if (cok) {
        const float bc = bias[col];
        #pragma unroll
        for (int i = 0; i < 8; ++i) {
            const int orow = base_row + i + 8 * half;
            out[(size_t)orow * C + col] = acc[i] + bc;
        }
    }
}

// ---------------------------------------------------------------- per-node log_softmax over 40 classes
__global__ void logsoftmax_kernel(const float* __restrict__ h, float* __restrict__ out) {
    int i = blockIdx.x * blockDim.x + threadIdx.x;
    if (i >= N_NODES) return;
    float v[C];
    float m = -1e30f;
    #pragma unroll
    for (int j = 0; j < C; ++j) { v[j] = h[(size_t)i * C + j]; m = fmaxf(m, v[j]); }
    float s = 0.0f;
    #pragma unroll
    for (int j = 0; j < C; ++j) s += __expf(v[j] - m);
    const float ls = __logf(s);
    #pragma unroll
    for (int j = 0; j < C; ++j) out[(size_t)i * C + j] = v[j] - m - ls;
}

// ----------------------------------------------------------------
extern "C" void kernel_launch(void* const* d_in, const int* in_sizes, int n_in,
                              void* d_out, int out_size, void* d_ws, size_t ws_size,
                              hipStream_t stream) {
    const float*     x     = (const float*)d_in[0];
    const long long* ei    = (const long long*)d_in[1];
    const float*     wl0   = (const float*)d_in[2];
    const float*     wr0   = (const float*)d_in[3];
    const float*     b0    = (const float*)d_in[4];
    const float*     wl1   = (const float*)d_in[5];
    const float*     wr1   = (const float*)d_in[6];
    const float*     b1    = (const float*)d_in[7];
    const float*     gamma = (const float*)d_in[8];
    const float*     beta  = (const float*)d_in[9];
    const float*     rmean = (const float*)d_in[10];
    const float*     rvar  = (const float*)d_in[11];
    const float*     wl2   = (const float*)d_in[12];
    const float*     wr2   = (const float*)d_in[13];
    const float*     b2    = (const float*)d_in[14];

    // output slices, in reference return order
    float* lsm  = (float*)d_out;                       // [N, C]  log_softmax
    float* act0 = lsm  + (size_t)N_NODES * C;          // [N, D]  == h0
    float* act1 = act0 + (size_t)N_NODES * D;          // [N, D]  == h1
    float* hout = act1 + (size_t)N_NODES * D;          // [N, C]  final conv

    // workspace: degrees + message accumulator + transposed weights
    float* deg  = (float*)d_ws;
    float* msg  = deg + (((size_t)N_NODES + 255) / 256) * 256;
    float* wl0T = msg  + (size_t)N_NODES * D;   // [128][128]
    float* wr0T = wl0T + D * D;
    float* wl1T = wr0T + D * D;
    float* wr1T = wl1T + D * D;
    float* wl2T = wr1T + D * D;                 // [40][128]
    float* wr2T = wl2T + C * D;
    const size_t msgN = (size_t)N_NODES * D;

    // transpose weights (tiny, amortized)
    transpose_kernel<<<(D * D + 255) / 256, 256, 0, stream>>>(wl0, wl0T, D, D);
    transpose_kernel<<<(D * D + 255) / 256, 256, 0, stream>>>(wr0, wr0T, D, D);
    transpose_kernel<<<(D * D + 255) / 256, 256, 0, stream>>>(wl1, wl1T, D, D);
    transpose_kernel<<<(D * D + 255) / 256, 256, 0, stream>>>(wr1, wr1T, D, D);
    transpose_kernel<<<(D * C + 255) / 256, 256, 0, stream>>>(wl2, wl2T, D, C);
    transpose_kernel<<<(D * C + 255) / 256, 256, 0, stream>>>(wr2, wr2T, D, C);

    // degrees -> reciprocal
    zero_kernel<<<1024, 256, 0, stream>>>(deg, (size_t)N_NODES);
    degree_kernel<<<(N_EDGES + 255) / 256, 256, 0, stream>>>(ei, deg);
    invdeg_kernel<<<(N_NODES + 255) / 256, 256, 0, stream>>>(deg);

    // ---- layer 0: h0 = relu(mean_agg(x) @ wl0 + b0 + x @ wr0)  -> act0
    zero_kernel<<<2048, 256, 0, stream>>>(msg, msgN);
    scatter_kernel<<<N_EDGES / 8, 256, 0, stream>>>(ei, x, msg);
    gemm128_kernel<<<N_NODES / 16, 256, 0, stream>>>(msg, deg, x, wl0T, wr0T, b0,
                                                     gamma, beta, rmean, rvar, x,
                                                     act0, 0);

    // ---- layer 1: h1 = relu(BN(mean_agg(h0) @ wl1 + b1 + h0 @ wr1)) + x -> act1
    zero_kernel<<<2048, 256, 0, stream>>>(msg, msgN);
    scatter_kernel<<<N_EDGES / 8, 256, 0, stream>>>(ei, act0, msg);
    gemm128_kernel<<<N_NODES / 16, 256, 0, stream>>>(msg, deg, act0, wl1T, wr1T, b1,
                                                     gamma, beta, rmean, rvar, x,
                                                     act1, 1);

    // ---- layer 2: h = mean_agg(h1) @ wl2 + b2 + h1 @ wr2 -> hout
    zero_kernel<<<2048, 256, 0, stream>>>(msg, msgN);
    scatter_kernel<<<N_EDGES / 8, 256, 0, stream>>>(ei, act1, msg);
    gemm_out_kernel<<<N_NODES / 16, 96, 0, stream>>>(msg, deg, act1, wl2T, wr2T, b2, hout);

    // ---- log_softmax(h) -> lsm
    logsoftmax_kernel<<<(N_NODES + 255) / 256, 256, 0, stream>>>(hout, lsm);
}